// VQLocalProbAvgPool_50027779064365
// MI455X (gfx1250) — compile-verified
//
#include <hip/hip_runtime.h>

// Problem constants (from the reference)
#define VQ      320
#define BB      16
#define LL      4096
#define DD      768
#define NSPLIT  16            // L-splits in the reduction kernel
#define CHUNK   (LL / NSPLIT) // 256 l-values per block
#define DTILES  6             // 6 * 128 = 768 d per batch
#define DPB     128           // d-values per block (8 waves * 16)

typedef __attribute__((ext_vector_type(2))) float v2f;
typedef __attribute__((ext_vector_type(8))) float v8f;

// ---------------------------------------------------------------------------
// Kernel 1: per-batch histograms -> weights w[l] = 1/(fx+fy), and invS = 1/Σw
// One block per batch. LDS histogram (ds_add atomics), deterministic tree
// reduction for the sum.
// ---------------------------------------------------------------------------
__global__ __launch_bounds__(256) void hist_weights_kernel(
    const long long* __restrict__ vq, float* __restrict__ W,
    float* __restrict__ invS) {
  __shared__ int   cx[VQ];
  __shared__ int   cy[VQ];
  __shared__ float red[256];
  const int b = blockIdx.x;
  const int t = threadIdx.x;

  for (int i = t; i < VQ; i += 256) { cx[i] = 0; cy[i] = 0; }
  __syncthreads();

  const long long* vb = vq + (size_t)b * LL * 2;
  for (int l = t; l < LL; l += 256) {
    atomicAdd(&cx[(int)vb[2 * l + 0]], 1);
    atomicAdd(&cy[(int)vb[2 * l + 1]], 1);
  }
  __syncthreads();

  float local = 0.0f;
  float* wb = W + (size_t)b * LL;
  for (int l = t; l < LL; l += 256) {
    const int fx = cx[(int)vb[2 * l + 0]];
    const int fy = cy[(int)vb[2 * l + 1]];
    const float w = 1.0f / (float)(fx + fy);  // softmax(log w) == w / Σw
    wb[l] = w;
    local += w;
  }
  red[t] = local;
  __syncthreads();
  for (int s = 128; s > 0; s >>= 1) {
    if (t < s) red[t] += red[t + s];
    __syncthreads();
  }
  if (t == 0) invS[b] = 1.0f / red[0];
}

// ---------------------------------------------------------------------------
// Kernel 2: partial[b,s,d] = Σ_{l in chunk s} w[b,l] * x[b,l,d]
// via V_WMMA_F32_16X16X4_F32:
//   A[m,k] = x[l0+k, dbase+m]   (documented 16x4 f32 A layout:
//             lane = {kk,m}: VGPR0 = K=2*kk, VGPR1 = K=2*kk+1, m = lane&15)
//   B[k,n] = w[l0+k] for n==0, else 0  -> D[m,0] accumulates the weighted sum.
// Each wave owns a 16-d tile and chains WMMAs over its 256-l chunk.
// ---------------------------------------------------------------------------
__global__ __launch_bounds__(256) void wsum_wmma_kernel(
    const float* __restrict__ x, const float* __restrict__ W,
    float* __restrict__ partial) {
  __shared__ float wlds[CHUNK];
  const int bid = blockIdx.x;
  const int b  = bid / (NSPLIT * DTILES);
  const int r  = bid % (NSPLIT * DTILES);
  const int s  = r / DTILES;
  const int dt = r % DTILES;
  const int t  = threadIdx.x;
  const int c0 = s * CHUNK;

  wlds[t] = W[(size_t)b * LL + c0 + t];  // CHUNK == blockDim.x == 256
  __syncthreads();

  // x = input_feature[:, -1] : skip the N=0 slab, never touch it (halves HBM)
  const float* xb = x + ((size_t)b * 2 + 1) * (size_t)LL * DD;

  const int wave = t >> 5;
  const int lane = t & 31;
  const int m    = lane & 15;   // M row (d offset) and also N column for B
  const int kk   = lane >> 4;   // selects K pair {2kk, 2kk+1}
  const int dbase = dt * DPB + wave * 16;
  const float sel = (m == 0) ? 1.0f : 0.0f;  // only column n==0 of B nonzero

  v8f acc0 = {};
  v8f acc1 = {};
  const float* xp = xb + (size_t)(c0 + 2 * kk) * DD + dbase + m;

#pragma unroll 4
  for (int li = 0; li < CHUNK; li += 8) {
    const float* p0 = xp + (size_t)li * DD;
    const float* p1 = p0 + (size_t)4 * DD;
    v2f a0, a1, b0, b1;
    a0.x = p0[0];
    a0.y = p0[DD];
    a1.x = p1[0];
    a1.y = p1[DD];
    b0.x = wlds[li + 2 * kk + 0] * sel;
    b0.y = wlds[li + 2 * kk + 1] * sel;
    b1.x = wlds[li + 4 + 2 * kk + 0] * sel;
    b1.y = wlds[li + 4 + 2 * kk + 1] * sel;
    acc0 = __builtin_amdgcn_wmma_f32_16x16x4_f32(false, a0, false, b0,
                                                 (short)0, acc0, false, false);
    acc1 = __builtin_amdgcn_wmma_f32_16x16x4_f32(false, a1, false, b1,
                                                 (short)0, acc1, false, false);
  }
  acc0 = acc0 + acc1;

  // D column n==0 lives in lane 0 (M=r, r=0..7) and lane 16 (M=8+r)
  if (m == 0) {
    float* pp = partial + (size_t)(b * NSPLIT + s) * DD + dbase + kk * 8;
#pragma unroll
    for (int q = 0; q < 8; ++q) pp[q] = acc0[q];
  }
}

// ---------------------------------------------------------------------------
// Kernel 3: out[b,d] = invS[b] * Σ_s partial[b,s,d]  (fixed-order, no atomics)
// ---------------------------------------------------------------------------
__global__ __launch_bounds__(256) void finalize_kernel(
    const float* __restrict__ partial, const float* __restrict__ invS,
    float* __restrict__ out) {
  const int i = blockIdx.x * 256 + threadIdx.x;
  if (i >= BB * DD) return;
  const int b = i / DD;
  const int d = i % DD;
  float acc = 0.0f;
#pragma unroll
  for (int k = 0; k < NSPLIT; ++k)
    acc += partial[(size_t)(b * NSPLIT + k) * DD + d];
  out[i] = acc * invS[b];
}

// ---------------------------------------------------------------------------
extern "C" void kernel_launch(void* const* d_in, const int* in_sizes, int n_in,
                              void* d_out, int out_size, void* d_ws,
                              size_t ws_size, hipStream_t stream) {
  (void)in_sizes; (void)n_in; (void)out_size; (void)ws_size;
  const float*     x   = (const float*)d_in[0];     // (16,2,4096,768) f32
  // d_in[1] = input_lengths : unused by the reference
  const long long* vq  = (const long long*)d_in[2]; // (16,4096,2) i64

  float* ws      = (float*)d_ws;
  float* W       = ws;                    // 16*4096 floats   (256 KiB)
  float* invS    = W + (size_t)BB * LL;   // 16 floats
  float* partial = invS + BB;             // 16*16*768 floats (768 KiB)

  hist_weights_kernel<<<BB, 256, 0, stream>>>(vq, W, invS);
  wsum_wmma_kernel<<<BB * NSPLIT * DTILES, 256, 0, stream>>>(x, W, partial);
  finalize_kernel<<<(BB * DD + 255) / 256, 256, 0, stream>>>(partial, invS,
                                                             (float*)d_out);
}